// PoseDepthUpdater_48799418417188
// MI455X (gfx1250) — compile-verified
//
#include <hip/hip_runtime.h>
#include <math.h>

// Fused pose + depth update for MI455X (gfx1250, wave32).
//
// Depth part is pure streaming (16 B/elem, ~0.125 flop/byte): bandwidth-bound,
// so we use float4 (global_load_b128 / global_store_b128) with a grid-stride
// pattern and 4 vec4 items per thread for MLP/ILP. Default RT cache hints keep
// the ~150 MB working set L2-resident (192 MB L2) across graph replays.
//
// Pose part (B=32 rows) rides along in block 0, lanes 0..31 (one wave32).

#define ILP 4
#define TPB 256

__global__ __launch_bounds__(TPB) void posedepth_fused(
    const float* __restrict__ curr_pose,    // [B,7]
    const float* __restrict__ curr_depth,   // [n] f32
    const float* __restrict__ delta_pose,   // [B,6]
    const float* __restrict__ delta_depth,  // [n] f32
    const float* __restrict__ a_p,          // [B]
    const float* __restrict__ a_d,          // [n] f32
    float* __restrict__ out_pose,           // [B,7]
    float* __restrict__ out_depth,          // [n] f32 (16B-aligned base)
    int Bn, int n)
{
    // ---------------- pose update: block 0, lanes 0..Bn-1 (one wave) -------
    if (blockIdx.x == 0 && (int)threadIdx.x < Bn) {
        const int b = threadIdx.x;
        const float s = a_p[b];
        const float vx = s * delta_pose[b * 6 + 0];
        const float vy = s * delta_pose[b * 6 + 1];
        const float vz = s * delta_pose[b * 6 + 2];
        const float wx = s * delta_pose[b * 6 + 3];
        const float wy = s * delta_pose[b * 6 + 4];
        const float wz = s * delta_pose[b * 6 + 5];

        const float th2 = wx * wx + wy * wy + wz * wz;
        const float th  = sqrtf(th2);
        const bool  is_small = th < 1e-4f;

        const float s_over = is_small ? (0.5f - th2 * (1.0f / 48.0f))
                                      : (sinf(0.5f * th) / th);
        const float qdx = s_over * wx;
        const float qdy = s_over * wy;
        const float qdz = s_over * wz;
        const float qdw = cosf(0.5f * th);

        const float Bc = is_small ? (0.5f - th2 * (1.0f / 24.0f))
                                  : ((1.0f - cosf(th)) / th2);
        const float Cc = is_small ? ((1.0f / 6.0f) - th2 * (1.0f / 120.0f))
                                  : ((th - sinf(th)) / (th2 * th));

        // c = w x v
        const float cx = wy * vz - wz * vy;
        const float cy = wz * vx - wx * vz;
        const float cz = wx * vy - wy * vx;
        // d = w x (w x v)
        const float dx = wy * cz - wz * cy;
        const float dy = wz * cx - wx * cz;
        const float dz = wx * cy - wy * cx;

        const float tdx = vx + Bc * cx + Cc * dx;
        const float tdy = vy + Bc * cy + Cc * dy;
        const float tdz = vz + Bc * cz + Cc * dz;

        const float tcx = curr_pose[b * 7 + 0];
        const float tcy = curr_pose[b * 7 + 1];
        const float tcz = curr_pose[b * 7 + 2];
        const float qcx = curr_pose[b * 7 + 3];
        const float qcy = curr_pose[b * 7 + 4];
        const float qcz = curr_pose[b * 7 + 5];
        const float qcw = curr_pose[b * 7 + 6];

        // qrot(q_d, t_c) = t + 2*(qdw*(u x t) + u x (u x t)), u = q_d.xyz
        const float ux = qdy * tcz - qdz * tcy;
        const float uy = qdz * tcx - qdx * tcz;
        const float uz = qdx * tcy - qdy * tcx;
        const float rx = tcx + 2.0f * (qdw * ux + (qdy * uz - qdz * uy));
        const float ry = tcy + 2.0f * (qdw * uy + (qdz * ux - qdx * uz));
        const float rz = tcz + 2.0f * (qdw * uz + (qdx * uy - qdy * ux));

        out_pose[b * 7 + 0] = rx + tdx;
        out_pose[b * 7 + 1] = ry + tdy;
        out_pose[b * 7 + 2] = rz + tdz;
        // qmul(q_d, q_c), xyzw Hamilton
        out_pose[b * 7 + 3] = qdw * qcx + qdx * qcw + qdy * qcz - qdz * qcy;
        out_pose[b * 7 + 4] = qdw * qcy - qdx * qcz + qdy * qcw + qdz * qcx;
        out_pose[b * 7 + 5] = qdw * qcz + qdx * qcy - qdy * qcx + qdz * qcw;
        out_pose[b * 7 + 6] = qdw * qcw - qdx * qcx - qdy * qcy - qdz * qcz;
    }

    // ---------------- depth update: vec4 streaming ------------------------
    const int n4     = n >> 2;
    const int stride = gridDim.x * blockDim.x;
    int j = blockIdx.x * blockDim.x + threadIdx.x;

    const float4* __restrict__ cd4 = reinterpret_cast<const float4*>(curr_depth);
    const float4* __restrict__ dd4 = reinterpret_cast<const float4*>(delta_depth);
    const float4* __restrict__ ad4 = reinterpret_cast<const float4*>(a_d);
    float4* __restrict__       od4 = reinterpret_cast<float4*>(out_depth);

#pragma unroll
    for (int k = 0; k < ILP; ++k, j += stride) {
        if (j < n4) {
            const float4 c = cd4[j];
            const float4 d = dd4[j];
            const float4 a = ad4[j];
            float4 r;
            r.x = fmaxf(fmaf(a.x, d.x, c.x), 0.1f);
            r.y = fmaxf(fmaf(a.y, d.y, c.y), 0.1f);
            r.z = fmaxf(fmaf(a.z, d.z, c.z), 0.1f);
            r.w = fmaxf(fmaf(a.w, d.w, c.w), 0.1f);
            od4[j] = r;
        }
    }

    // Scalar tail (n % 4 != 0 never happens for W=640, kept for robustness).
    if (blockIdx.x == 0 && threadIdx.x == 0) {
        for (int t = n4 << 2; t < n; ++t) {
            out_depth[t] = fmaxf(fmaf(a_d[t], delta_depth[t], curr_depth[t]), 0.1f);
        }
    }
}

extern "C" void kernel_launch(void* const* d_in, const int* in_sizes, int n_in,
                              void* d_out, int out_size, void* d_ws, size_t ws_size,
                              hipStream_t stream) {
    // setup_inputs() order:
    // 0: curr_pose [B,7], 1: curr_depth [B,H,W], 2: delta_pose [B,6],
    // 3: delta_depth [B,H,W], 4: a_p [B,1], 5: a_d [B,H,W]  (all float32)
    const float* curr_pose   = (const float*)d_in[0];
    const float* curr_depth  = (const float*)d_in[1];
    const float* delta_pose  = (const float*)d_in[2];
    const float* delta_depth = (const float*)d_in[3];
    const float* a_p         = (const float*)d_in[4];
    const float* a_d         = (const float*)d_in[5];

    const int Bn = in_sizes[4];        // B (a_p has B elements)
    const int n  = in_sizes[1];        // B*H*W

    // Output: new_pose [B,7] first, then new_depth [B*H*W].
    float* out_pose  = (float*)d_out;
    float* out_depth = out_pose + (size_t)Bn * 7;  // byte offset B*28 (=896 for B=32): 16B aligned

    const int n4 = n >> 2;
    long threads_needed = ((long)n4 + ILP - 1) / ILP;
    int blocks = (int)((threads_needed + TPB - 1) / TPB);
    if (blocks < 1) blocks = 1;

    posedepth_fused<<<blocks, TPB, 0, stream>>>(
        curr_pose, curr_depth, delta_pose, delta_depth, a_p, a_d,
        out_pose, out_depth, Bn, n);
}